// S4DKernel_10393820856907
// MI455X (gfx1250) — compile-verified
//
#include <hip/hip_runtime.h>

typedef __attribute__((ext_vector_type(2))) float v2f;
typedef __attribute__((ext_vector_type(8))) float v8f;

namespace {
constexpr int kH = 512;          // d_model
constexpr int kN = 32;           // complex modes per channel (= wave32 lanes)
constexpr int kL = 8192;         // kernel length
constexpr int kChunks = 32;      // L-chunks per channel (parallelism + drift control)
constexpr int kChunkL = kL / kChunks;   // 256 l's per wave
constexpr int kRoundL = 32;             // l's per LDS round (2 WMMA chains)
constexpr int kRounds = kChunkL / kRoundL;  // 8
constexpr int kLdsStride = 33;          // pad to dodge bank conflicts on transpose
}

// One wave (32 lanes) per block. Lane n owns complex mode n of channel h.
// State S[n] = 2 * Cs[n] * w[n]^l advances by S *= w each l. Every 32 steps the
// real parts are transposed through LDS and reduced over n with two independent
// chains of 8 accumulating V_WMMA_F32_16X16X4_F32 ops (A = ones). Because all D
// rows replicate, lanes 0-15 carry chain0's 16 outputs and lanes 16-31 carry
// chain1's, giving one full-wave coalesced 128B store per round.
__global__ __launch_bounds__(32) void s4d_vandermonde_wmma(
    const float* __restrict__ log_dt,
    const float* __restrict__ log_A_real,
    const float* __restrict__ A_imag,
    const float* __restrict__ Cm,      // (H, N, 2) real view of complex C
    float* __restrict__ out)           // (H, L)
{
  __shared__ float lds[kRoundL * kLdsStride];

  const int lane  = threadIdx.x;            // 0..31 == mode index n
  const int h     = blockIdx.x / kChunks;
  const int chunk = blockIdx.x - h * kChunks;
  const int l0    = chunk * kChunkL;
  const int hn    = h * kN + lane;

  // --- per-mode setup (transcendentals: O(H*N*chunks), not O(H*N*L)) ---
  const float dt   = expf(log_dt[h]);
  const float Ar   = -expf(log_A_real[hn]);
  const float Ai   = A_imag[hn];
  const float dtAr = Ar * dt;
  const float dtAi = Ai * dt;

  // w = exp(dtA)
  const float er = expf(dtAr);
  const float wr = er * cosf(dtAi);
  const float wi = er * sinf(dtAi);

  // Cs = C * (exp(dtA) - 1) / A, with the final factor 2 folded in
  const float nr  = wr - 1.0f;
  const float ni  = wi;
  const float inv = 1.0f / (Ar * Ar + Ai * Ai);
  const float fr  = (nr * Ar + ni * Ai) * inv;
  const float fi  = (ni * Ar - nr * Ai) * inv;
  const float Cre = Cm[2 * hn + 0];
  const float Cim = Cm[2 * hn + 1];
  const float Csr = 2.0f * (Cre * fr - Cim * fi);
  const float Csi = 2.0f * (Cre * fi + Cim * fr);

  // Seed state at l0: S = 2*Cs * exp(dtA * l0)
  const float t0 = (float)l0;
  const float e0 = expf(dtAr * t0);
  const float c0 = cosf(dtAi * t0);
  const float s0 = sinf(dtAi * t0);
  const float Er = e0 * c0;
  const float Ei = e0 * s0;
  float Sr = Csr * Er - Csi * Ei;
  float Si = Csr * Ei + Csi * Er;

  // B-operand gather pattern for V_WMMA_F32_16X16X4_F32 (B is 4x16, K x N):
  //   VGPR0: lanes 0-15 -> K=0 (N=lane), lanes 16-31 -> K=2 (N=lane-16)
  //   VGPR1: lanes 0-15 -> K=1,          lanes 16-31 -> K=3
  const int r  = lane & 15;                  // output column j (= l within 16-tile)
  const int kb = (lane < 16) ? 0 : 2;        // K-row pair select
  const v2f onesA = {1.0f, 1.0f};            // A = ones(16x4): pure reduction

  float* outp = out + (size_t)h * kL + l0;

  for (int round = 0; round < kRounds; ++round) {
    // 32 recurrence steps; stash real parts [l_in_round][n] in LDS
#pragma unroll
    for (int s = 0; s < kRoundL; ++s) {
      lds[s * kLdsStride + lane] = Sr;
      const float tr = Sr * wr - Si * wi;
      const float ti = Sr * wi + Si * wr;
      Sr = tr;
      Si = ti;
    }
    __syncthreads();

    // Two independent reductions over the 32 modes (8 f32 WMMAs each, K=4)
    v8f acc0 = {0.f, 0.f, 0.f, 0.f, 0.f, 0.f, 0.f, 0.f};  // l_in_round 0..15
    v8f acc1 = {0.f, 0.f, 0.f, 0.f, 0.f, 0.f, 0.f, 0.f};  // l_in_round 16..31
#pragma unroll
    for (int s = 0; s < 8; ++s) {
      const float* bp0 = &lds[r * kLdsStride + 4 * s + kb];
      const float* bp1 = &lds[(r + 16) * kLdsStride + 4 * s + kb];
      v2f B0, B1;
      B0.x = bp0[0];
      B0.y = bp0[1];
      B1.x = bp1[0];
      B1.y = bp1[1];
      acc0 = __builtin_amdgcn_wmma_f32_16x16x4_f32(
          false, onesA, false, B0, (short)0, acc0, false, false);
      acc1 = __builtin_amdgcn_wmma_f32_16x16x4_f32(
          false, onesA, false, B1, (short)0, acc1, false, false);
    }

    // D rows replicate (A = ones): lane j<16 holds chain0 column j in acc0[0];
    // lane j>=16 holds chain1 column j-16 in acc1[0] (row M=8 duplicate).
    // -> one unconditional full-wave coalesced store of 32 consecutive floats.
    const float val = (lane < 16) ? acc0[0] : acc1[0];
    outp[round * kRoundL + lane] = val;

    __syncthreads();  // protect LDS before next round's scan overwrites it
  }
}

extern "C" void kernel_launch(void* const* d_in, const int* in_sizes, int n_in,
                              void* d_out, int out_size, void* d_ws, size_t ws_size,
                              hipStream_t stream) {
  (void)in_sizes; (void)n_in; (void)d_ws; (void)ws_size; (void)out_size;
  const float* log_dt     = (const float*)d_in[0];
  const float* log_A_real = (const float*)d_in[1];
  const float* A_imag     = (const float*)d_in[2];
  const float* Cm         = (const float*)d_in[3];
  // d_in[4] is the Python scalar L (= 8192), baked in at compile time.
  float* out = (float*)d_out;

  s4d_vandermonde_wmma<<<dim3(kH * kChunks), dim3(32), 0, stream>>>(
      log_dt, log_A_real, A_imag, Cm, out);
}